// YUV420ChannelLoss_50483045597471
// MI455X (gfx1250) — compile-verified
//
#include <hip/hip_runtime.h>

// ---------------------------------------------------------------------------
// YUV420 channel loss, MI455X (gfx1250), wave32.
// Pipeline: diff(RGB->dY,dU,dV) -> separable 15-tap box-sum via
// V_WMMA_F32_16X16X4_F32 band matmuls -> sum of squares -> scalar.
//
// Roofline: 100 MB input read (HBM) + 25 MB intermediate (L2-resident,
// 192 MB L2) => ~5.4 us floor at 23.3 TB/s. Compute is negligible; fp32
// WMMA keeps full precision vs the fp32 JAX reference.
// ---------------------------------------------------------------------------

typedef __attribute__((ext_vector_type(2))) float v2f;
typedef __attribute__((ext_vector_type(8))) float v8f;

__device__ __forceinline__ v8f wmma_f32_16x16x4(v2f a, v2f b, v8f c) {
  // 8-arg pattern: (neg_a, A, neg_b, B, c_mod, C, reuse_a, reuse_b)
  return __builtin_amdgcn_wmma_f32_16x16x4_f32(false, a, false, b, (short)0, c,
                                               false, false);
}

// ---------------------------------------------------------------------------
// Kernel 1: per-pixel YUV difference. One thread = one 2x4 pixel strip
// (two 2x2 chroma quads). All global accesses are b128/b64, fully coalesced.
// dY full-res, dU/dV quarter-res (2x2 mean). +128 chroma offsets cancel.
// ---------------------------------------------------------------------------
__global__ __launch_bounds__(256) void yuv_diff_kernel(
    const float* __restrict__ in, const float* __restrict__ tg,
    float* __restrict__ dY, float* __restrict__ dU, float* __restrict__ dV) {
  const int t = blockIdx.x * 256 + threadIdx.x;  // 16*256*128 threads total
  const int n = t >> 15;
  const int qy = (t >> 7) & 255;   // chroma row
  const int qx = t & 127;          // strip index: luma cols 4qx..4qx+3
  const size_t base = (size_t)n * 3 * 262144;    // 262144 = 512*512

  // Per-quad RGB-difference sums (quad0 = cols 4qx..4qx+1, quad1 = +2..+3)
  float sdr0 = 0.f, sdg0 = 0.f, sdb0 = 0.f;
  float sdr1 = 0.f, sdg1 = 0.f, sdb1 = 0.f;
#pragma unroll
  for (int dr_ = 0; dr_ < 2; ++dr_) {
    const int row = 2 * qy + dr_;
    const size_t rb = base + (size_t)row * 512;
    const float4 ri = reinterpret_cast<const float4*>(in + rb)[qx];
    const float4 gi = reinterpret_cast<const float4*>(in + rb + 262144)[qx];
    const float4 bi = reinterpret_cast<const float4*>(in + rb + 524288)[qx];
    const float4 rt = reinterpret_cast<const float4*>(tg + rb)[qx];
    const float4 gt = reinterpret_cast<const float4*>(tg + rb + 262144)[qx];
    const float4 bt = reinterpret_cast<const float4*>(tg + rb + 524288)[qx];
    const float4 dr = ri - rt;
    const float4 dg = gi - gt;
    const float4 db = bi - bt;
    float4 dy4;
    dy4.x = 0.299f * dr.x + 0.587f * dg.x + 0.114f * db.x;
    dy4.y = 0.299f * dr.y + 0.587f * dg.y + 0.114f * db.y;
    dy4.z = 0.299f * dr.z + 0.587f * dg.z + 0.114f * db.z;
    dy4.w = 0.299f * dr.w + 0.587f * dg.w + 0.114f * db.w;
    reinterpret_cast<float4*>(dY + ((size_t)n * 512 + row) * 512)[qx] = dy4;
    sdr0 += dr.x + dr.y;  sdg0 += dg.x + dg.y;  sdb0 += db.x + db.y;
    sdr1 += dr.z + dr.w;  sdg1 += dg.z + dg.w;  sdb1 += db.z + db.w;
  }
  const size_t crow = ((size_t)n * 256 + qy) * 256;
  float2 u2, v2;
  u2.x = 0.25f * (-0.169f * sdr0 - 0.331f * sdg0 + 0.5f * sdb0);
  u2.y = 0.25f * (-0.169f * sdr1 - 0.331f * sdg1 + 0.5f * sdb1);
  v2.x = 0.25f * (0.5f * sdr0 - 0.46f * sdg0 - 0.04f * sdb0);
  v2.y = 0.25f * (0.5f * sdr1 - 0.46f * sdg1 - 0.04f * sdb1);
  reinterpret_cast<float2*>(dU + crow)[qx] = u2;
  reinterpret_cast<float2*>(dV + crow)[qx] = v2;
}

// ---------------------------------------------------------------------------
// Kernel 2: per 16x16 output tile (one wave32), 15x15 box-sum via two banded
// fp32 WMMA matmuls, then sum-of-squares -> atomicAdd into accumulator.
//
// fp32 WMMA fragment layouts (ISA 7.12.2), lane L: lo=L&15, hi=L>>4
//   A  16x4 : M = lo, vgpr v holds K = v + 2*hi          (within K=4 chunk)
//   B  4x16 : N = lo, vgpr v holds K = v + 2*hi          (symmetric)
//   C/D16x16: element r <-> M = r + 8*hi, N = lo
//
// The band fragment band[kk] = [K in [lo, lo+14]] is identical for the
// vertical-pass A operand and the horizontal-pass B operand: compute once.
// ---------------------------------------------------------------------------
__global__ __launch_bounds__(32) void boxloss_kernel(
    const float* __restrict__ field, int H, int W, float* __restrict__ accum) {
  __shared__ float Dl[32][33];  // 30x30 halo'd diff tile, rows/cols 30,31 = 0
  __shared__ float Tl[16][33];  // vertical-pass result, cols 30,31 = 0

  const int lane = threadIdx.x;
  const int hi = lane >> 4, lo = lane & 15;
  const int row0 = blockIdx.y * 16, col0 = blockIdx.x * 16;
  const float* img = field + (size_t)blockIdx.z * H * W;

  // Banded-ones fragment, pure function of (lane, kk): 1 if K in [lo, lo+14].
  v2f band[8];
#pragma unroll
  for (int kk = 0; kk < 8; ++kk) {
    const int K = kk * 4 + hi * 2;
    band[kk].x = ((unsigned)(K - lo) < 15u) ? 1.f : 0.f;
    band[kk].y = ((unsigned)(K + 1 - lo) < 15u) ? 1.f : 0.f;
  }

  // Cooperative load of the 30x30 region (zero-padded) into 32x32 LDS block.
  for (int idx = lane; idx < 32 * 32; idx += 32) {
    const int r = idx >> 5, c = idx & 31;
    const int gr = row0 - 7 + r, gc = col0 - 7 + c;
    float v = 0.f;
    if (r < 30 && c < 30 && (unsigned)gr < (unsigned)H &&
        (unsigned)gc < (unsigned)W)
      v = img[(size_t)gr * W + gc];
    Dl[r][c] = v;
  }
  Tl[lo][30 + hi] = 0.f;  // zero the K=30,31 columns of T
  __syncthreads();

  // ---- vertical pass: T[i][c] = sum_{j=i..i+14} Dl[j][c], c in 0..29 ----
  v8f t0 = {}, t1 = {};
#pragma unroll
  for (int kk = 0; kk < 8; ++kk) {
    const int K = kk * 4 + hi * 2;
    v2f b0, b1;
    b0.x = Dl[K][lo];
    b0.y = Dl[K + 1][lo];
    b1.x = Dl[K][14 + lo];
    b1.y = Dl[K + 1][14 + lo];
    t0 = wmma_f32_16x16x4(band[kk], b0, t0);  // T cols 0..15
    t1 = wmma_f32_16x16x4(band[kk], b1, t1);  // T cols 14..29
  }
#pragma unroll
  for (int r = 0; r < 8; ++r) {
    Tl[r + 8 * hi][lo] = t0[r];
    if (lo >= 2) Tl[r + 8 * hi][14 + lo] = t1[r];  // avoid overlap race
  }
  __syncthreads();

  // ---- horizontal pass: F[i][c] = sum_{j=c..c+14} T[i][j] ----
  v8f f = {};
#pragma unroll
  for (int kk = 0; kk < 8; ++kk) {
    const int K = kk * 4 + hi * 2;
    v2f a;
    a.x = Tl[lo][K];
    a.y = Tl[lo][K + 1];
    f = wmma_f32_16x16x4(a, band[kk], f);
  }

  // ---- sum of squares over this lane's 8 outputs, wave32 reduce ----
  float s = 0.f;
#pragma unroll
  for (int r = 0; r < 8; ++r) s += f[r] * f[r];
#pragma unroll
  for (int off = 16; off >= 1; off >>= 1) s += __shfl_xor(s, off, 32);
  if (lane == 0) atomicAdd(accum, s);
}

// ---------------------------------------------------------------------------
__global__ void init_kernel(float* __restrict__ accum) {
  if (threadIdx.x < 4) accum[threadIdx.x] = 0.f;
}

__global__ void finalize_kernel(const float* __restrict__ accum,
                                float* __restrict__ out) {
  const float invY = 1.f / (16.f * 512.f * 512.f);
  const float invC = 1.f / (16.f * 256.f * 256.f);
  out[0] = accum[0] * invY + (accum[1] + accum[2]) * invC;
}

// ---------------------------------------------------------------------------
extern "C" void kernel_launch(void* const* d_in, const int* in_sizes, int n_in,
                              void* d_out, int out_size, void* d_ws,
                              size_t ws_size, hipStream_t stream) {
  (void)in_sizes; (void)n_in; (void)out_size; (void)ws_size;
  const float* input = (const float*)d_in[0];
  const float* target = (const float*)d_in[1];
  float* out = (float*)d_out;

  float* ws = (float*)d_ws;
  float* accum = ws;                          // 16 floats (3 used)
  float* dY = ws + 16;                        // 16*512*512
  float* dU = dY + 16 * 512 * 512;            // 16*256*256
  float* dV = dU + 16 * 256 * 256;            // 16*256*256

  init_kernel<<<1, 32, 0, stream>>>(accum);

  // 16*256*128 strips, 256 threads/block
  yuv_diff_kernel<<<2048, 256, 0, stream>>>(input, target, dY, dU, dV);

  boxloss_kernel<<<dim3(32, 32, 16), 32, 0, stream>>>(dY, 512, 512, accum + 0);
  boxloss_kernel<<<dim3(16, 16, 16), 32, 0, stream>>>(dU, 256, 256, accum + 1);
  boxloss_kernel<<<dim3(16, 16, 16), 32, 0, stream>>>(dV, 256, 256, accum + 2);

  finalize_kernel<<<1, 1, 0, stream>>>(accum, out);
}